// PointTransformerLayer_40819369181267
// MI455X (gfx1250) — compile-verified
//
#include <hip/hip_runtime.h>
#include <hip/hip_bf16.h>

// Problem constants (from reference): B,N,C,O,K = 2, 8192, 64, 64, 16
#define B_   2
#define N_   8192
#define C_   64
#define O_   64
#define K_   16
#define NK   17          // top-(K+1), slot 0 is "self"
#define EPSF 1e-5f
#define FLT_MAX_ 3.402823466e+38f

typedef float v2f __attribute__((ext_vector_type(2)));
typedef float v8f __attribute__((ext_vector_type(8)));

#if defined(__has_builtin)
#if __has_builtin(__builtin_amdgcn_wmma_f32_16x16x4_f32)
#define HAVE_WMMA_F32 1
#endif
#endif

__device__ __forceinline__ v8f v8f_zero() {
  v8f z = {0.f, 0.f, 0.f, 0.f, 0.f, 0.f, 0.f, 0.f};
  return z;
}

// D = A(16x4 f32) * B(4x16 f32) + C(16x16 f32), wave32 distributed.
// A frag (per lane): m = lane&15 is the row; lanes<16 hold k=0,1; lanes>=16 hold k=2,3.
// B frag (per lane): col = lane&15; lanes<16 hold k=0,1; lanes>=16 hold k=2,3.
// C/D: VGPR r -> row r (lanes<16) / row r+8 (lanes>=16); col = lane&15.
__device__ __forceinline__ v8f wmma_f32_16x16x4(v2f a, v2f b, v8f c) {
#ifdef HAVE_WMMA_F32
  return __builtin_amdgcn_wmma_f32_16x16x4_f32(
      /*neg_a=*/false, a, /*neg_b=*/false, b,
      /*c_mod=*/(short)0, c, /*reuse_a=*/false, /*reuse_b=*/false);
#else
  (void)a; (void)b;
  return c;
#endif
}

// Swizzled-destination index for staging a 64x64 row-major weight into LDS in
// B-fragment order: frag (s,t) for lane l lives at ((s*4+t)*32 + l)*2 + {0,1}.
// Given a linear source index i (= krow*64 + col), return the LDS index.
__device__ __forceinline__ int bfrag_dst(int i) {
  const int krow = i >> 6;          // 0..63
  const int col  = i & 63;          // 0..63
  const int s    = krow >> 2;       // k-step
  const int kr   = krow & 3;        // k within step
  const int j    = kr & 1;          // element within the lane's pair
  const int khi  = kr >> 1;         // lane half
  const int t    = col >> 4;        // col tile
  const int m    = col & 15;        // col within tile
  return (((s * 4 + t) * 32) + (khi * 16 + m)) * 2 + j;
}

// ---------------------------------------------------------------------------
// Kernel A: Q/K/V projections.  X (B*N x 64) @ W (64x64) + b, three weights.
// Block = 256 threads = 8 waves; each wave computes a 16-row x 64-col tile for
// all three outputs, reusing one A fragment per k-step for 12 WMMAs.
// Weights are staged into LDS pre-swizzled so each B fragment is ONE
// ds_load_b64 directly into an even-aligned VGPR pair (no repack movs).
// ---------------------------------------------------------------------------
__global__ __launch_bounds__(256) void qkv_proj_kernel(
    const float* __restrict__ x,
    const float* __restrict__ wq, const float* __restrict__ bq,
    const float* __restrict__ wk, const float* __restrict__ bk,
    const float* __restrict__ wv, const float* __restrict__ bv,
    float* __restrict__ xq, float* __restrict__ xk, float* __restrict__ xv)
{
  __shared__ __align__(16) float wS[3][C_ * O_];   // 3 * 16 KB, B-frag order
  __shared__ float bS[3][O_];

  const int tid = threadIdx.x;
  for (int i = tid; i < C_ * O_; i += 256) {
    const int d = bfrag_dst(i);
    wS[0][d] = wq[i];
    wS[1][d] = wk[i];
    wS[2][d] = wv[i];
  }
  if (tid < O_) { bS[0][tid] = bq[tid]; bS[1][tid] = bk[tid]; bS[2][tid] = bv[tid]; }
  __syncthreads();

  const int wave = tid >> 5;
  const int lane = tid & 31;
  const int m    = lane & 15;
  const int half = lane >> 4;
  const int rowBase = (blockIdx.x * 8 + wave) * 16;

  v8f acc[3][4];
#pragma unroll
  for (int g = 0; g < 3; ++g)
#pragma unroll
    for (int t = 0; t < 4; ++t) acc[g][t] = v8f_zero();

  // A row for this lane; fragments are contiguous f32 pairs -> global_load_b64
  const v2f* xrow2 = (const v2f*)(x + (size_t)(rowBase + m) * C_);
  const v2f* wf[3] = {(const v2f*)wS[0], (const v2f*)wS[1], (const v2f*)wS[2]};

#pragma unroll
  for (int s = 0; s < 16; ++s) {          // 16 k-steps of 4 -> K = 64
    const v2f a = xrow2[2 * s + half];
#pragma unroll
    for (int g = 0; g < 3; ++g) {
#pragma unroll
      for (int t = 0; t < 4; ++t) {
        const v2f bb = wf[g][(s * 4 + t) * 32 + lane];   // one ds_load_b64
        acc[g][t] = wmma_f32_16x16x4(a, bb, acc[g][t]);
      }
    }
  }

  float* outs[3] = {xq, xk, xv};
#pragma unroll
  for (int g = 0; g < 3; ++g) {
#pragma unroll
    for (int t = 0; t < 4; ++t) {
#pragma unroll
      for (int r = 0; r < 8; ++r) {
        const int row = rowBase + r + 8 * half;
        const int col = t * 16 + m;
        outs[g][(size_t)row * O_ + col] = acc[g][t][r] + bS[g][col];
      }
    }
  }
}

// ---------------------------------------------------------------------------
// Kernel B: brute-force kNN (top-17 smallest d2, ties by lower index; slot 0
// is the self point and is dropped on output, matching top_k(-d2, K+1)[1:]).
// Thread = one query; candidates streamed through LDS in 256-point tiles.
// ---------------------------------------------------------------------------
#define QT 256
#define CT 256
__global__ __launch_bounds__(QT) void knn_kernel(
    const float* __restrict__ p, int* __restrict__ ind)
{
  __shared__ float4 cand[CT];

  const int b = blockIdx.y;
  const int q = blockIdx.x * QT + threadIdx.x;
  const float* pb = p + (size_t)b * N_ * 3;

  const float px = pb[q * 3 + 0];
  const float py = pb[q * 3 + 1];
  const float pz = pb[q * 3 + 2];
  const float sq = px * px + py * py + pz * pz;

  float bd[NK];
  int   bi[NK];
#pragma unroll
  for (int i = 0; i < NK; ++i) { bd[i] = FLT_MAX_; bi[i] = 0x7fffffff; }

  for (int c0 = 0; c0 < N_; c0 += CT) {
    __syncthreads();
    {
      const int j = threadIdx.x;   // QT == CT
      const float cx = pb[(c0 + j) * 3 + 0];
      const float cy = pb[(c0 + j) * 3 + 1];
      const float cz = pb[(c0 + j) * 3 + 2];
      cand[j] = make_float4(cx, cy, cz, cx * cx + cy * cy + cz * cz);
    }
    __syncthreads();

    if (c0 + CT < N_)  // speculative prefetch of next candidate tile -> global_prefetch_b8
      __builtin_prefetch(pb + (size_t)(c0 + CT + threadIdx.x) * 3, 0, 0);

#pragma unroll 4
    for (int j = 0; j < CT; ++j) {
      const float4 cv = cand[j];
      const float d2 = sq + cv.w - 2.0f * (px * cv.x + py * cv.y + pz * cv.z);
      const int   id = c0 + j;
      // Guard: only do the (rare) insertion when we beat the current worst.
      if (d2 < bd[NK - 1] || (d2 == bd[NK - 1] && id < bi[NK - 1])) {
        // Branch-free shift-insert, fully unrolled (stays in VGPRs).
#pragma unroll
        for (int t = NK - 1; t >= 0; --t) {
          const bool bt  = (d2 < bd[t]) || (d2 == bd[t] && id < bi[t]);
          const bool bt1 = (t > 0) && ((d2 < bd[t - 1]) || (d2 == bd[t - 1] && id < bi[t - 1]));
          if (bt) {
            bd[t] = bt1 ? bd[t - 1] : d2;
            bi[t] = bt1 ? bi[t - 1] : id;
          }
        }
      }
    }
  }

  int* op = ind + ((size_t)b * N_ + q) * K_;
#pragma unroll
  for (int k = 0; k < K_; ++k) op[k] = bi[k + 1];   // drop self
}

// ---------------------------------------------------------------------------
// Kernel C: fused attention.  One wave per query point.
//  pr = relu(bn_p((pj - p) @ lp1 + b)) @ lp2 + b          (tiny, per-lane)
//  w0 = relu(bn1(xq - gather(xk) + pr))                   -> LDS 16x64 tile
//  w1 = relu(bn2(w0 @ lw1 + b))   via WMMA f32 16x16x4    -> LDS round trip
//  w2 = w1 @ lw2 + b              via WMMA
//  softmax over K (rows) per column; out = sum_k (xv+pr)*w2
// ---------------------------------------------------------------------------
#define WPB 4
__global__ __launch_bounds__(WPB * 32) void attn_kernel(
    const float* __restrict__ p,
    const float* __restrict__ xq, const float* __restrict__ xk,
    const float* __restrict__ xv, const int* __restrict__ ind,
    const float* __restrict__ lp1_w, const float* __restrict__ lp1_b,
    const float* __restrict__ bnp_g, const float* __restrict__ bnp_b,
    const float* __restrict__ bnp_m, const float* __restrict__ bnp_v,
    const float* __restrict__ lp2_w, const float* __restrict__ lp2_b,
    const float* __restrict__ bn1_g, const float* __restrict__ bn1_b,
    const float* __restrict__ bn1_m, const float* __restrict__ bn1_v,
    const float* __restrict__ lw1_w, const float* __restrict__ lw1_b,
    const float* __restrict__ bn2_g, const float* __restrict__ bn2_b,
    const float* __restrict__ bn2_m, const float* __restrict__ bn2_v,
    const float* __restrict__ lw2_w, const float* __restrict__ lw2_b,
    float* __restrict__ out)
{
  __shared__ __align__(16) float w1S[O_ * O_];      // 16 KB, B-frag order
  __shared__ __align__(16) float w2S[O_ * O_];      // 16 KB, B-frag order
  __shared__ float s1S[O_], t1S[O_], s2S[O_], t2S[O_], lb1S[O_], lb2S[O_];
  __shared__ __align__(16) float prS[WPB][K_ * O_]; // 16 KB
  __shared__ __align__(16) float wS [WPB][K_ * O_]; // 16 KB
  __shared__ int   nbrS[WPB][K_];

  const int tid = threadIdx.x;
  for (int i = tid; i < O_ * O_; i += WPB * 32) {
    const int d = bfrag_dst(i);
    w1S[d] = lw1_w[i];
    w2S[d] = lw2_w[i];
  }
  if (tid < O_) {
    const float s1 = bn1_g[tid] * rsqrtf(bn1_v[tid] + EPSF);
    s1S[tid] = s1; t1S[tid] = bn1_b[tid] - bn1_m[tid] * s1;
    const float s2 = bn2_g[tid] * rsqrtf(bn2_v[tid] + EPSF);
    s2S[tid] = s2; t2S[tid] = bn2_b[tid] - bn2_m[tid] * s2;
    lb1S[tid] = lw1_b[tid]; lb2S[tid] = lw2_b[tid];
  }
  __syncthreads();

  const int wave = tid >> 5;
  const int lane = tid & 31;
  const int gq   = blockIdx.x * WPB + wave;       // 0 .. B*N-1
  const int b    = gq / N_;
  const int n    = gq % N_;
  const int kl   = lane & 15;
  const int half = lane >> 4;

  // --- neighbors for this query ---
  if (lane < K_) nbrS[wave][lane] = ind[(size_t)gq * K_ + lane];
  __builtin_amdgcn_wave_barrier();   // intra-wave LDS is in-order on CDNA5

  // --- position-encoding MLP: lane -> (neighbor kl, channels half*32..+31) ---
  {
    const float* pq = p + ((size_t)b * N_ + n) * 3;
    const int    nb = nbrS[wave][kl];
    const float* pj = p + ((size_t)b * N_ + nb) * 3;
    const float r0 = pj[0] - pq[0], r1 = pj[1] - pq[1], r2 = pj[2] - pq[2];
    float h[3];
#pragma unroll
    for (int c = 0; c < 3; ++c) {
      float v = r0 * lp1_w[0 * 3 + c] + r1 * lp1_w[1 * 3 + c] + r2 * lp1_w[2 * 3 + c] + lp1_b[c];
      v = (v - bnp_m[c]) * rsqrtf(bnp_v[c] + EPSF) * bnp_g[c] + bnp_b[c];
      h[c] = fmaxf(v, 0.f);
    }
#pragma unroll
    for (int c = 0; c < 32; ++c) {
      const int cc = half * 32 + c;
      prS[wave][kl * O_ + cc] =
          h[0] * lp2_w[0 * O_ + cc] + h[1] * lp2_w[1 * O_ + cc] + h[2] * lp2_w[2 * O_ + cc] + lp2_b[cc];
    }
  }
  __builtin_amdgcn_wave_barrier();

  // --- w0 = relu(bn1(xq - xk[nbr] + pr)) -> wS tile (16 x 64) ---
  {
    const float* xqr = xq + (size_t)gq * O_;
    const int    nb  = nbrS[wave][kl];
    const float* xkr = xk + ((size_t)b * N_ + nb) * O_;
#pragma unroll
    for (int c = 0; c < 32; ++c) {
      const int cc = half * 32 + c;
      float v = xqr[cc] - xkr[cc] + prS[wave][kl * O_ + cc];
      wS[wave][kl * O_ + cc] = fmaxf(v * s1S[cc] + t1S[cc], 0.f);
    }
  }
  __builtin_amdgcn_wave_barrier();

  const v2f* w1f = (const v2f*)w1S;
  const v2f* w2f = (const v2f*)w2S;
  const v2f* wSf = (const v2f*)wS[wave];   // A frags: contiguous pairs, b64 loads

  // --- GEMM1: (16x64) @ lw1 (64x64), epilogue bn2+relu, back to wS ---
  {
    v8f acc[4];
#pragma unroll
    for (int t = 0; t < 4; ++t) acc[t] = v8f_zero();
#pragma unroll
    for (int s = 0; s < 16; ++s) {
      const v2f a = wSf[kl * 32 + 2 * s + half];         // wS[kl][4s+kk .. +1]
#pragma unroll
      for (int t = 0; t < 4; ++t) {
        const v2f bb = w1f[(s * 4 + t) * 32 + lane];     // one ds_load_b64
        acc[t] = wmma_f32_16x16x4(a, bb, acc[t]);
      }
    }
    __builtin_amdgcn_wave_barrier();   // all A-reads done before overwrite
#pragma unroll
    for (int t = 0; t < 4; ++t) {
#pragma unroll
      for (int r = 0; r < 8; ++r) {
        const int row = r + 8 * half;
        const int cc  = t * 16 + kl;
        float v = acc[t][r] + lb1S[cc];
        wS[wave][row * O_ + cc] = fmaxf(v * s2S[cc] + t2S[cc], 0.f);
      }
    }
  }
  __builtin_amdgcn_wave_barrier();

  // --- GEMM2: (16x64) @ lw2 (64x64) kept in registers; softmax over rows ---
  {
    v8f acc[4];
#pragma unroll
    for (int t = 0; t < 4; ++t) acc[t] = v8f_zero();
#pragma unroll
    for (int s = 0; s < 16; ++s) {
      const v2f a = wSf[kl * 32 + 2 * s + half];
#pragma unroll
      for (int t = 0; t < 4; ++t) {
        const v2f bb = w2f[(s * 4 + t) * 32 + lane];
        acc[t] = wmma_f32_16x16x4(a, bb, acc[t]);
      }
    }

#pragma unroll
    for (int t = 0; t < 4; ++t) {
      const int cc = t * 16 + kl;
      // bias + row-max (rows split: this lane has rows r+8*half; lane^16 the rest)
      float mx = -FLT_MAX_;
#pragma unroll
      for (int r = 0; r < 8; ++r) { acc[t][r] += lb2S[cc]; mx = fmaxf(mx, acc[t][r]); }
      mx = fmaxf(mx, __shfl_xor(mx, 16, 32));
      float sum = 0.f;
#pragma unroll
      for (int r = 0; r < 8; ++r) {
        const float ev = __expf(acc[t][r] - mx);
        acc[t][r] = ev;
        sum += ev;
      }
      sum += __shfl_xor(sum, 16, 32);
      const float inv = 1.0f / sum;

      float po = 0.f;
#pragma unroll
      for (int r = 0; r < 8; ++r) {
        const int row = r + 8 * half;
        const int nb  = nbrS[wave][row];
        const float xvv = xv[((size_t)b * N_ + nb) * O_ + cc];
        po += (xvv + prS[wave][row * O_ + cc]) * (acc[t][r] * inv);
      }
      po += __shfl_xor(po, 16, 32);
      if (half == 0) out[(size_t)gq * O_ + cc] = po;
    }
  }
}

// ---------------------------------------------------------------------------
extern "C" void kernel_launch(void* const* d_in, const int* in_sizes, int n_in,
                              void* d_out, int out_size, void* d_ws, size_t ws_size,
                              hipStream_t stream) {
  (void)in_sizes; (void)n_in; (void)out_size; (void)ws_size;

  const float* x     = (const float*)d_in[0];
  const float* p     = (const float*)d_in[1];
  const float* wq    = (const float*)d_in[2];
  const float* bq    = (const float*)d_in[3];
  const float* wk    = (const float*)d_in[4];
  const float* bk    = (const float*)d_in[5];
  const float* wv    = (const float*)d_in[6];
  const float* bv    = (const float*)d_in[7];
  const float* lp1_w = (const float*)d_in[8];
  const float* lp1_b = (const float*)d_in[9];
  const float* bnp_g = (const float*)d_in[10];
  const float* bnp_b = (const float*)d_in[11];
  const float* bnp_m = (const float*)d_in[12];
  const float* bnp_v = (const float*)d_in[13];
  const float* lp2_w = (const float*)d_in[14];
  const float* lp2_b = (const float*)d_in[15];
  const float* bn1_g = (const float*)d_in[16];
  const float* bn1_b = (const float*)d_in[17];
  const float* bn1_m = (const float*)d_in[18];
  const float* bn1_v = (const float*)d_in[19];
  const float* lw1_w = (const float*)d_in[20];
  const float* lw1_b = (const float*)d_in[21];
  const float* bn2_g = (const float*)d_in[22];
  const float* bn2_b = (const float*)d_in[23];
  const float* bn2_m = (const float*)d_in[24];
  const float* bn2_v = (const float*)d_in[25];
  const float* lw2_w = (const float*)d_in[26];
  const float* lw2_b = (const float*)d_in[27];

  // Workspace: Q, K, V projections (3 x B*N*O floats = 12 MB)
  float* xq = (float*)d_ws;
  float* xk = xq + (size_t)B_ * N_ * O_;
  float* xv = xk + (size_t)B_ * N_ * O_;

  // Output layout: out (B*N*O f32) then ind (B*N*K i32 bits)
  float* out = (float*)d_out;
  int*   ind = (int*)(out + (size_t)B_ * N_ * O_);

  // A: QKV projections (WMMA f32)
  qkv_proj_kernel<<<dim3((B_ * N_) / 128), 256, 0, stream>>>(
      x, wq, bq, wk, bk, wv, bv, xq, xk, xv);

  // B: kNN indices
  knn_kernel<<<dim3(N_ / QT, B_), QT, 0, stream>>>(p, ind);

  // C: fused attention (WMMA f32), one wave per query
  attn_kernel<<<dim3((B_ * N_) / WPB), WPB * 32, 0, stream>>>(
      p, xq, xk, xv, ind,
      lp1_w, lp1_b, bnp_g, bnp_b, bnp_m, bnp_v, lp2_w, lp2_b,
      bn1_g, bn1_b, bn1_m, bn1_v, lw1_w, lw1_b,
      bn2_g, bn2_b, bn2_m, bn2_v, lw2_w, lw2_b,
      out);
}